// DegreeGCNPlusLayer_27642409517695
// MI455X (gfx1250) — compile-verified
//
#include <hip/hip_runtime.h>

// Problem constants (from the reference)
#define N_NODES 100000
#define N_EDGES 640000
#define D       128     // D_IN == D_OUT

typedef __attribute__((ext_vector_type(2))) float v2f;
typedef __attribute__((ext_vector_type(8))) float v8f;

// ---------------------------------------------------------------------------
// Kernel 1: zero the h accumulator (N_NODES * D floats in d_ws).
// N_NODES*D/4 = 3,200,000 float4 -> 12500 blocks of 256, exact.
// ---------------------------------------------------------------------------
__global__ __launch_bounds__(256) void zero_h(float4* __restrict__ h) {
    h[(size_t)blockIdx.x * blockDim.x + threadIdx.x] = make_float4(0.f, 0.f, 0.f, 0.f);
}

// ---------------------------------------------------------------------------
// Kernel 2: edge scatter-add.  One wave32 per edge; lane l handles 4
// consecutive floats (float4 load -> 4 global_atomic_add_f32).
// Gather reads are fully coalesced (512 B per wave, one source row).
// Accumulator h (51.2 MB) is L2-resident so atomics resolve in L2.
// ---------------------------------------------------------------------------
__global__ __launch_bounds__(256) void scatter_add(const float* __restrict__ x,
                                                   const int*   __restrict__ src,
                                                   const int*   __restrict__ dst,
                                                   float*       __restrict__ h) {
    const int tid  = blockIdx.x * blockDim.x + threadIdx.x;
    const int e    = tid >> 5;
    const int lane = tid & 31;
    if (e >= N_EDGES) return;
    const int s = src[e];
    const int d = dst[e];
    const float4 v = ((const float4*)(x + (size_t)s * D))[lane];
    float* hr = h + (size_t)d * D + lane * 4;
    atomicAdd(hr + 0, v.x);
    atomicAdd(hr + 1, v.y);
    atomicAdd(hr + 2, v.z);
    atomicAdd(hr + 3, v.w);
}

// ---------------------------------------------------------------------------
// Kernel 3: out = (h / degree[:,None]) @ W + b  via V_WMMA_F32_16X16X4_F32.
//
// Block = 256 threads = 8 wave32s. Block b owns rows [16*b, 16*b+16);
// wave w owns output cols [16*w, 16*w+16).  K = 128 -> 32 unrolled wmma ops.
//
// ISA VGPR layouts (CDNA5 ISA):
//  A (16x4 f32): lanes 0-15 hold row M=lane, K = k+0..k+1 in 2 VGPRs;
//                lanes 16-31 hold row M=lane-16, K = k+2..k+3.
//  B (4x16 f32): row-striped across lanes: VGPR j, lanes 0-15 -> K=k+2*half+j,
//                N=lane; (half = lane>>4 selects K group, mirroring A).
//  C/D (16x16):  VGPR v, lanes 0-15 -> M=v, N=lane; lanes 16-31 -> M=8+v.
//
// Degree normalization is fused into the A-fragment load.
// ---------------------------------------------------------------------------
__global__ __launch_bounds__(256) void gemm_wmma(const float* __restrict__ h,
                                                 const float* __restrict__ degree,
                                                 const float* __restrict__ W,
                                                 const float* __restrict__ b,
                                                 float*       __restrict__ out) {
    const int wave  = threadIdx.x >> 5;   // 0..7 -> n-tile
    const int lane  = threadIdx.x & 31;
    const int lhalf = lane >> 4;          // 0 or 1
    const int l15   = lane & 15;

    const int m0 = blockIdx.x * 16;       // 6250 blocks * 16 = 100000 exact
    const int n0 = wave * 16;             // 8 waves * 16 = 128 = D_OUT exact

    // A fragment source: row m0+l15, columns k + 2*lhalf + {0,1}
    const int    arow    = m0 + l15;
    const float  inv_deg = 1.0f / degree[arow];
    const float* aptr    = h + (size_t)arow * D + 2 * lhalf;

    // B fragment source: col n0+l15, rows k + 2*lhalf + {0,1}
    const float* bptr = W + (size_t)(2 * lhalf) * D + n0 + l15;

    v8f acc = {};
#pragma unroll
    for (int k = 0; k < D; k += 4) {
        v2f a = *(const v2f*)(aptr + k);          // 8B-aligned (row base 512B-aligned)
        a.x *= inv_deg;
        a.y *= inv_deg;
        v2f bb;
        bb.x = bptr[(size_t)(k + 0) * D];
        bb.y = bptr[(size_t)(k + 1) * D];
        // 8 args: (neg_a, A, neg_b, B, c_mod, C, reuse_a, reuse_b)
        acc = __builtin_amdgcn_wmma_f32_16x16x4_f32(
            false, a, false, bb, (short)0, acc, false, false);
    }

    // Bias + store per C/D layout: VGPR v -> row m0 + 8*lhalf + v, col n0+l15.
    const float bias = b[n0 + l15];
    float* orow = out + (size_t)(m0 + lhalf * 8) * D + n0 + l15;
#pragma unroll
    for (int v = 0; v < 8; ++v) {
        orow[(size_t)v * D] = acc[v] + bias;
    }
}

// ---------------------------------------------------------------------------
// Harness entry point.
// Input order (setup_inputs dict): inputs, src, dst, degree, W, b
// d_ws holds the f32 accumulator h: N_NODES * D * 4 = 51.2 MB.
// ---------------------------------------------------------------------------
extern "C" void kernel_launch(void* const* d_in, const int* in_sizes, int n_in,
                              void* d_out, int out_size, void* d_ws, size_t ws_size,
                              hipStream_t stream) {
    const float* inputs = (const float*)d_in[0];
    const int*   src    = (const int*)  d_in[1];
    const int*   dst    = (const int*)  d_in[2];
    const float* degree = (const float*)d_in[3];
    const float* W      = (const float*)d_in[4];
    const float* bias   = (const float*)d_in[5];
    float*       out    = (float*)d_out;
    float*       h      = (float*)d_ws;

    (void)in_sizes; (void)n_in; (void)out_size; (void)ws_size;

    // 1) h = 0
    zero_h<<<(N_NODES * D / 4) / 256, 256, 0, stream>>>((float4*)h);
    // 2) h[dst] += inputs[src]   (one wave per edge)
    scatter_add<<<(N_EDGES * 32) / 256, 256, 0, stream>>>(inputs, src, dst, h);
    // 3) out = (h/deg) @ W + b   (WMMA f32 16x16x4)
    gemm_wmma<<<N_NODES / 16, 256, 0, stream>>>(h, degree, W, bias, out);
}